// DiffeomorphicTransform_34857954574416
// MI455X (gfx1250) — compile-verified
//
#include <hip/hip_runtime.h>
#include <stdint.h>

// ---------------------------------------------------------------------------
// DiffeomorphicTransform (scaling & squaring), 160x192x160 x 3ch f32.
//   flow0 = velocity / 32
//   5x:  flow <- flow + trilerp_clamped(flow, sample_pos(flow))
// sample_pos(i,j,k) = (i + f0*79.5, j + f1*95.5, k + f2*79.5)   [z,y,x]
//
// Optimization notes (MI455X):
//  * Bandwidth-bound: 2x59MB ping-pong buffers fit in 192MB L2 -> all 5
//    passes after the first HBM touch run L2-resident (~23.3TB/s HBM is only
//    hit once for velocity + final writeback).
//  * First pass fused with the /32 init (trilerp is linear; *2^-5 is exact),
//    removing one full 59MB read + 59MB write pass (6 passes -> 5).
//  * One block per (i,j) row: k = tid, i,j from uniform SALU div; no
//    per-thread integer div/mod chains.
//  * CDNA5 async path: each block stages its contiguous 1920B own-flow row
//    into LDS via global_load_async_to_lds_b128 (120 lanes x 16B), completed
//    with s_wait_asynccnt + barrier, overlapped with index setup.
//  * No WMMA: per-voxel gather/interp has no shared-operand matmul
//    structure; matrix pipes cannot help a bandwidth-bound resampler.
// ---------------------------------------------------------------------------

#define DD 160
#define HH 192
#define WW 160
#define NP (DD * HH * WW)        /* 4,915,200 voxels */
#define NROW (DD * HH)           /* 30,720 rows of WW voxels */

__device__ __forceinline__ void gather_fma(const float* __restrict__ s,
                                           int x, int y, int z, float w,
                                           float& a0, float& a1, float& a2) {
  const float* p = s + 3u * (uint32_t)((z * HH + y) * WW + x);  // b96 gather
  a0 = fmaf(w, p[0], a0);
  a1 = fmaf(w, p[1], a1);
  a2 = fmaf(w, p[2], a2);
}

// One squaring step: dst = f + trilerp(src_field, pos(f)).
// FIRST=true: src holds raw velocity; f = src*2^-5 and the gathered sum is
// scaled by 2^-5 at the end (bit-exact vs. scaling src up front).
template <bool FIRST>
__global__ __launch_bounds__(WW) void squaring_step(const float* __restrict__ src,
                                                    float* __restrict__ dst) {
  __shared__ float own[WW * 3];                 // 1920 B: this row's 3ch flow
  const int t   = threadIdx.x;                  // k coordinate
  const int row = blockIdx.x;                   // flat (i,j)

  // --- async DMA: stage this row's contiguous 1920B of src into LDS.
  // 120 lanes x b128. Flat LDS pointer low 32 bits are the wave-relative
  // LDS byte address (flat-LDS aperture mapping).
  if (t < (WW * 3) / 4) {
    const float* g = src + (size_t)row * (WW * 3) + t * 4;
    uint32_t loff = (uint32_t)(uintptr_t)(&own[t * 4]);
    uint64_t ga   = (uint64_t)(uintptr_t)g;
    asm volatile("global_load_async_to_lds_b128 %0, %1, off"
                 :: "v"(loff), "v"(ga) : "memory");
  }

  // Uniform (SALU) index setup overlaps the DMA.
  const int j = row % HH;
  const int i = row / HH;
  const int k = t;

  asm volatile("s_wait_asynccnt 0" ::: "memory");
  __syncthreads();   // lanes read bytes staged by other waves' DMA

  float f0 = own[3 * t + 0];   // D-direction channel
  float f1 = own[3 * t + 1];   // H-direction channel
  float f2 = own[3 * t + 2];   // W-direction channel
  if (FIRST) { f0 *= 0.03125f; f1 *= 0.03125f; f2 *= 0.03125f; }

  // Sample position in voxel units (clamped-border trilinear).
  const float x = (float)k + f2 * 79.5f;   // (WW-1)/2
  const float y = (float)j + f1 * 95.5f;   // (HH-1)/2
  const float z = (float)i + f0 * 79.5f;   // (DD-1)/2

  const float xf = floorf(x), yf = floorf(y), zf = floorf(z);
  const float fx = x - xf, fy = y - yf, fz = z - zf;

  const int x0 = (int)xf, y0 = (int)yf, z0 = (int)zf;
  const int x0c = x0 < 0 ? 0 : (x0 > WW - 1 ? WW - 1 : x0);
  const int x1c = x0 + 1 < 0 ? 0 : (x0 + 1 > WW - 1 ? WW - 1 : x0 + 1);
  const int y0c = y0 < 0 ? 0 : (y0 > HH - 1 ? HH - 1 : y0);
  const int y1c = y0 + 1 < 0 ? 0 : (y0 + 1 > HH - 1 ? HH - 1 : y0 + 1);
  const int z0c = z0 < 0 ? 0 : (z0 > DD - 1 ? DD - 1 : z0);
  const int z1c = z0 + 1 < 0 ? 0 : (z0 + 1 > DD - 1 ? DD - 1 : z0 + 1);

  const float wx0 = 1.0f - fx, wx1 = fx;
  const float wy0 = 1.0f - fy, wy1 = fy;
  const float wz0 = 1.0f - fz, wz1 = fz;

  float a0 = 0.0f, a1 = 0.0f, a2 = 0.0f;
  // Same corner order as the reference (dx fastest, then dy, then dz).
  gather_fma(src, x0c, y0c, z0c, (wx0 * wy0) * wz0, a0, a1, a2);
  gather_fma(src, x1c, y0c, z0c, (wx1 * wy0) * wz0, a0, a1, a2);
  gather_fma(src, x0c, y1c, z0c, (wx0 * wy1) * wz0, a0, a1, a2);
  gather_fma(src, x1c, y1c, z0c, (wx1 * wy1) * wz0, a0, a1, a2);
  gather_fma(src, x0c, y0c, z1c, (wx0 * wy0) * wz1, a0, a1, a2);
  gather_fma(src, x1c, y0c, z1c, (wx1 * wy0) * wz1, a0, a1, a2);
  gather_fma(src, x0c, y1c, z1c, (wx0 * wy1) * wz1, a0, a1, a2);
  gather_fma(src, x1c, y1c, z1c, (wx1 * wy1) * wz1, a0, a1, a2);

  if (FIRST) { a0 *= 0.03125f; a1 *= 0.03125f; a2 *= 0.03125f; }

  float* o = dst + 3u * (uint32_t)(row * WW + t);
  o[0] = f0 + a0;
  o[1] = f1 + a1;
  o[2] = f2 + a2;
}

extern "C" void kernel_launch(void* const* d_in, const int* in_sizes, int n_in,
                              void* d_out, int out_size, void* d_ws, size_t ws_size,
                              hipStream_t stream) {
  (void)in_sizes; (void)n_in; (void)out_size; (void)ws_size;

  const float* vel = (const float*)d_in[0];   // (NP, 3) f32, never mutated
  float* B = (float*)d_out;                   // ping-pong buffer B (final)
  float* A = (float*)d_ws;                    // needs NP*3*4 = 59 MB scratch

  // 5 squaring steps; fused first step reads raw velocity.
  // vel -> B -> A -> B -> A -> B : flow5 lands in d_out.
  squaring_step<true ><<<NROW, WW, 0, stream>>>(vel, B);
  squaring_step<false><<<NROW, WW, 0, stream>>>(B, A);
  squaring_step<false><<<NROW, WW, 0, stream>>>(A, B);
  squaring_step<false><<<NROW, WW, 0, stream>>>(B, A);
  squaring_step<false><<<NROW, WW, 0, stream>>>(A, B);
}